// ModelLink_81088982548495
// MI455X (gfx1250) — compile-verified
//
#include <hip/hip_runtime.h>
#include <hip/hip_bf16.h>

typedef float v2f __attribute__((ext_vector_type(2)));
typedef float v8f __attribute__((ext_vector_type(8)));

#define H      128
#define HH     (128 * 128)
#define NWRD   200000
#define NENT   50000
#define NSEN   20000
#define NE_WE  400000
#define NE_EW  400000
#define NE_WS  800000
#define NE_SW  800000
#define NE_LBL 200000

// ---------------------------------------------------------------- utilities
__global__ void zero_f32(float* __restrict__ p, long n) {
    long i = (long)blockIdx.x * blockDim.x + threadIdx.x;
    long stride = (long)gridDim.x * blockDim.x;
    for (; i < n; i += stride) p[i] = 0.0f;
}

// xw[i,h] = word_x[i,:23] @ Wlin[:,h] + b[h] + word_emb[wid[i],h]
__global__ void compute_xw(const float* __restrict__ wx,
                           const float* __restrict__ Wlin,
                           const float* __restrict__ b,
                           const float* __restrict__ wemb,
                           const int* __restrict__ wid,
                           float* __restrict__ out) {
    long idx = (long)blockIdx.x * blockDim.x + threadIdx.x;
    if (idx >= (long)NWRD * H) return;
    int i = (int)(idx >> 7), h = (int)(idx & 127);
    float s = b[h];
#pragma unroll
    for (int k = 0; k < 23; ++k)
        s = fmaf(wx[(long)i * 23 + k], Wlin[k * H + h], s);
    s += wemb[(long)wid[i] * H + h];
    out[idx] = s;
}

// out[i,:] = src[id[i],:]
__global__ void gather_rows(const float* __restrict__ src,
                            const int* __restrict__ id,
                            float* __restrict__ out, int n) {
    long idx = (long)blockIdx.x * blockDim.x + threadIdx.x;
    if (idx >= (long)n * H) return;
    int i = (int)(idx >> 7), h = (int)(idx & 127);
    out[idx] = src[(long)id[i] * H + h];
}

// accum[dst,:] += xsrc[src,:]; cnt[dst] += 1   (32 lanes per edge, float4 each)
__global__ void scatter_sum(const float* __restrict__ xsrc,
                            const int* __restrict__ es,
                            const int* __restrict__ ed,
                            float* __restrict__ acc,
                            float* __restrict__ cnt, int E) {
    long tid = (long)blockIdx.x * blockDim.x + threadIdx.x;
    int e = (int)(tid >> 5);
    if (e >= E) return;
    int c = (int)(tid & 31);
    int s = es[e], d = ed[e];
    float4 v = ((const float4*)(xsrc + (long)s * H))[c];
    float* dp = acc + (long)d * H + c * 4;
    atomicAdd(dp + 0, v.x);
    atomicAdd(dp + 1, v.y);
    atomicAdd(dp + 2, v.z);
    atomicAdd(dp + 3, v.w);
    if (c == 0) atomicAdd(cnt + d, 1.0f);
}

// -------------------------------------------------- fused SAGE output GEMM
// One wave computes a 32x128 output tile (two 16-row strips -> B frags reused
// twice). fp32 WMMA 16x16x4, K=128 per part, up to 4 parts:
//   out = relu?( sum_p (Cp ? Ap/max(Cp,1) : Ap) @ Wp + b0 + b1 )
#define ASTRIDE 132  // 128+4 floats: distinct LDS banks across the 16 A rows

__device__ __forceinline__ void gemm_part(const float* __restrict__ A,
                                          const float* __restrict__ C,
                                          const float* __restrict__ W,
                                          float* As, v8f accL[8], v8f accU[8],
                                          int m0, int nmax, int lane, int l16,
                                          int half) {
    __builtin_prefetch(W, 0, 0);  // global_prefetch_b8 hint on the weights
    // stage 32x128 A rows into LDS, fusing the segment-mean divide; rows past
    // the matrix end (wave-uniform tail) clamp to the last valid row
    for (int i = lane; i < 32 * 32; i += 32) {
        int r = i >> 5, c4 = i & 31;
        int gr = m0 + r;
        gr = (gr < nmax) ? gr : (nmax - 1);
        float4 v = ((const float4*)(A + (long)gr * H))[c4];
        if (C) {
            float sc = 1.0f / fmaxf(C[gr], 1.0f);
            v.x *= sc; v.y *= sc; v.z *= sc; v.w *= sc;
        }
        *((float4*)(As + r * ASTRIDE + c4 * 4)) = v;
    }
    asm volatile("s_wait_dscnt 0" ::: "memory");  // LDS RAW within the wave

    for (int kb = 0; kb < 32; ++kb) {
        int k = kb * 4 + half * 2;
        // A frags (ISA 7.12.2): lane = M row, half-wave selects K pair
        v2f aL, aU;
        aL.x = As[l16 * ASTRIDE + k];
        aL.y = As[l16 * ASTRIDE + k + 1];
        aU.x = As[(l16 + 16) * ASTRIDE + k];
        aU.y = As[(l16 + 16) * ASTRIDE + k + 1];
#pragma unroll
        for (int t = 0; t < 8; ++t) {
            // B frag: lane = N col, half-wave selects K pair (row-major W)
            v2f b;
            b.x = W[(long)k * H + t * 16 + l16];
            b.y = W[(long)(k + 1) * H + t * 16 + l16];
            accL[t] = __builtin_amdgcn_wmma_f32_16x16x4_f32(
                false, aL, false, b, (short)0, accL[t], false, false);
            accU[t] = __builtin_amdgcn_wmma_f32_16x16x4_f32(
                false, aU, false, b, (short)0, accU[t], false, false);
        }
    }
    asm volatile("s_wait_dscnt 0" ::: "memory");  // WAR vs next part's fill
}

__global__ __launch_bounds__(128) void sage_gemm_wmma(
    const float* __restrict__ A0, const float* __restrict__ C0, const float* __restrict__ W0,
    const float* __restrict__ A1, const float* __restrict__ C1, const float* __restrict__ W1,
    const float* __restrict__ A2, const float* __restrict__ C2, const float* __restrict__ W2,
    const float* __restrict__ A3, const float* __restrict__ C3, const float* __restrict__ W3,
    const float* __restrict__ b0, const float* __restrict__ b1,
    float* __restrict__ out, int nstrips, int nparts, int relu) {
    extern __shared__ float ldsbuf[];  // 4 waves x 32 x ASTRIDE floats

    int wave = threadIdx.x >> 5;
    int lane = threadIdx.x & 31;
    int pid = blockIdx.x * 4 + wave;             // 32-row pair index
    int npairs = (nstrips + 1) >> 1;
    if (pid >= npairs) return;                   // wave-uniform: EXEC all-1s
    int m0 = pid * 32;
    int nmax = nstrips * 16;
    int valid_hi = (m0 + 16) < nmax;             // wave-uniform

    int l16 = lane & 15;
    int half = lane >> 4;
    float* As = ldsbuf + wave * (32 * ASTRIDE);

    v8f accL[8], accU[8];
#pragma unroll
    for (int t = 0; t < 8; ++t) {
        float bv = (b0 ? b0[t * 16 + l16] : 0.0f) + (b1 ? b1[t * 16 + l16] : 0.0f);
        v8f z;
#pragma unroll
        for (int r = 0; r < 8; ++r) z[r] = bv;
        accL[t] = z;
        accU[t] = z;
    }

    // direct calls keep pointers traceable to kernel args -> global_load path
    gemm_part(A0, C0, W0, As, accL, accU, m0, nmax, lane, l16, half);
    if (nparts > 1) gemm_part(A1, C1, W1, As, accL, accU, m0, nmax, lane, l16, half);
    if (nparts > 2) {
        gemm_part(A2, C2, W2, As, accL, accU, m0, nmax, lane, l16, half);
        gemm_part(A3, C3, W3, As, accL, accU, m0, nmax, lane, l16, half);
    }

    // D layout: vgpr r -> row base + r + 8*half, lane -> col t*16 + l16
#pragma unroll
    for (int t = 0; t < 8; ++t)
#pragma unroll
        for (int r = 0; r < 8; ++r) {
            float v = accL[t][r];
            if (relu) v = fmaxf(v, 0.0f);
            out[(long)(m0 + r + 8 * half) * H + t * 16 + l16] = v;
        }
    if (valid_hi) {
#pragma unroll
        for (int t = 0; t < 8; ++t)
#pragma unroll
            for (int r = 0; r < 8; ++r) {
                float v = accU[t][r];
                if (relu) v = fmaxf(v, 0.0f);
                out[(long)(m0 + 16 + r + 8 * half) * H + t * 16 + l16] = v;
            }
    }
}

// pred[e] = dot(xw[l0[e],:], xe[l1[e],:])   one wave per edge
__global__ void edge_dot(const float* __restrict__ xw,
                         const float* __restrict__ xe,
                         const int* __restrict__ l0,
                         const int* __restrict__ l1,
                         float* __restrict__ out) {
    long tid = (long)blockIdx.x * blockDim.x + threadIdx.x;
    int e = (int)(tid >> 5);
    if (e >= NE_LBL) return;
    int lane = (int)(tid & 31);
    float4 a = ((const float4*)(xw + (long)l0[e] * H))[lane];
    float4 b = ((const float4*)(xe + (long)l1[e] * H))[lane];
    float s = a.x * b.x + a.y * b.y + a.z * b.z + a.w * b.w;
    for (int o = 16; o; o >>= 1) s += __shfl_xor(s, o, 32);
    if (lane == 0) out[e] = s;
}

// ---------------------------------------------------------------- launcher
extern "C" void kernel_launch(void* const* d_in, const int* in_sizes, int n_in,
                              void* d_out, int out_size, void* d_ws, size_t ws_size,
                              hipStream_t stream) {
    (void)in_sizes; (void)n_in; (void)out_size; (void)ws_size;

    const float* word_x = (const float*)d_in[0];
    const int*   wid    = (const int*)d_in[1];
    const int*   eid    = (const int*)d_in[2];
    const int*   sid    = (const int*)d_in[3];
    const int*   e_we   = (const int*)d_in[4];
    const int*   e_ew   = (const int*)d_in[5];
    const int*   e_ws   = (const int*)d_in[6];
    const int*   e_sw   = (const int*)d_in[7];
    const int*   e_lbl  = (const int*)d_in[8];
    const float* Wlin   = (const float*)d_in[9];
    const float* blin   = (const float*)d_in[10];
    const float* wemb   = (const float*)d_in[11];
    const float* eemb   = (const float*)d_in[12];
    const float* Wl1    = (const float*)d_in[13];
    const float* bl1    = (const float*)d_in[14];
    const float* Wr1    = (const float*)d_in[15];
    const float* Wl2    = (const float*)d_in[16];
    const float* bl2    = (const float*)d_in[17];
    const float* Wr2    = (const float*)d_in[18];
    float* pred = (float*)d_out;

    // workspace layout (floats)
    size_t off = 0;
    auto alloc = [&](size_t n) { float* p = (float*)d_ws + off; off += (n + 15) & ~(size_t)15; return p; };
    float* xw0 = alloc((size_t)NWRD * H);
    float* xe0 = alloc((size_t)NENT * H);
    float* xs0 = alloc((size_t)NSEN * H);
    float* xw1 = alloc((size_t)NWRD * H);
    float* xe1 = alloc((size_t)NENT * H);
    float* xs1 = alloc((size_t)NSEN * H);
    float* xw2 = alloc((size_t)NWRD * H);
    float* xe2 = alloc((size_t)NENT * H);
    // contiguous accumulator region: one zero pass covers everything
    float* zbase  = (float*)d_ws + off;
    float* acc_e  = alloc((size_t)NENT * H);
    float* acc_wa = alloc((size_t)NWRD * H);
    float* acc_wb = alloc((size_t)NWRD * H);
    float* acc_s  = alloc((size_t)NSEN * H);
    float* cnt_e  = alloc(NENT);
    float* cnt_wa = alloc(NWRD);
    float* cnt_wb = alloc(NWRD);
    float* cnt_s  = alloc(NSEN);
    long   ztotal = (long)(((float*)d_ws + off) - zbase);

    const int TB = 256;
    auto cdiv = [](long a, long b) { return (int)((a + b - 1) / b); };
    const size_t shmem = 4 * 32 * ASTRIDE * sizeof(float);  // 67.6 KB / block
    auto ggrid = [&](int nstrips) { return cdiv((nstrips + 1) / 2, 4); };

    // node features
    compute_xw<<<cdiv((long)NWRD * H, TB), TB, 0, stream>>>(word_x, Wlin, blin, wemb, wid, xw0);
    gather_rows<<<cdiv((long)NENT * H, TB), TB, 0, stream>>>(eemb, eid, xe0, NENT);
    gather_rows<<<cdiv((long)NSEN * H, TB), TB, 0, stream>>>(eemb, sid, xs0, NSEN);

    int se = NENT / 16, sw = NWRD / 16, ss = NSEN / 16;

    // -------- layer 1
    zero_f32<<<2048, TB, 0, stream>>>(zbase, ztotal);
    scatter_sum<<<cdiv((long)NE_WE * 32, TB), TB, 0, stream>>>(xw0, e_we, e_we + NE_WE, acc_e,  cnt_e,  NE_WE);
    scatter_sum<<<cdiv((long)NE_EW * 32, TB), TB, 0, stream>>>(xe0, e_ew, e_ew + NE_EW, acc_wa, cnt_wa, NE_EW);
    scatter_sum<<<cdiv((long)NE_SW * 32, TB), TB, 0, stream>>>(xs0, e_sw, e_sw + NE_SW, acc_wb, cnt_wb, NE_SW);
    scatter_sum<<<cdiv((long)NE_WS * 32, TB), TB, 0, stream>>>(xw0, e_ws, e_ws + NE_WS, acc_s,  cnt_s,  NE_WS);

    sage_gemm_wmma<<<ggrid(se), 128, shmem, stream>>>(
        acc_e, cnt_e, Wl1 + 0 * HH,  xe0, nullptr, Wr1 + 0 * HH,
        nullptr, nullptr, nullptr,   nullptr, nullptr, nullptr,
        bl1 + 0 * H, nullptr, xe1, se, 2, 1);
    sage_gemm_wmma<<<ggrid(sw), 128, shmem, stream>>>(
        acc_wa, cnt_wa, Wl1 + 1 * HH,  xw0, nullptr, Wr1 + 1 * HH,
        acc_wb, cnt_wb, Wl1 + 3 * HH,  xw0, nullptr, Wr1 + 3 * HH,
        bl1 + 1 * H, bl1 + 3 * H, xw1, sw, 4, 1);
    sage_gemm_wmma<<<ggrid(ss), 128, shmem, stream>>>(
        acc_s, cnt_s, Wl1 + 2 * HH,  xw0, nullptr, Wr1 + 2 * HH,
        nullptr, nullptr, nullptr,   nullptr, nullptr, nullptr,
        bl1 + 2 * H, nullptr, xs1, ss, 2, 1);

    // -------- layer 2 (out_s dead: skipped, along with its ws scatter)
    zero_f32<<<2048, TB, 0, stream>>>(zbase, ztotal);
    scatter_sum<<<cdiv((long)NE_WE * 32, TB), TB, 0, stream>>>(xw1, e_we, e_we + NE_WE, acc_e,  cnt_e,  NE_WE);
    scatter_sum<<<cdiv((long)NE_EW * 32, TB), TB, 0, stream>>>(xe1, e_ew, e_ew + NE_EW, acc_wa, cnt_wa, NE_EW);
    scatter_sum<<<cdiv((long)NE_SW * 32, TB), TB, 0, stream>>>(xs1, e_sw, e_sw + NE_SW, acc_wb, cnt_wb, NE_SW);

    sage_gemm_wmma<<<ggrid(se), 128, shmem, stream>>>(
        acc_e, cnt_e, Wl2 + 0 * HH,  xe1, nullptr, Wr2 + 0 * HH,
        nullptr, nullptr, nullptr,   nullptr, nullptr, nullptr,
        bl2 + 0 * H, nullptr, xe2, se, 2, 0);
    sage_gemm_wmma<<<ggrid(sw), 128, shmem, stream>>>(
        acc_wa, cnt_wa, Wl2 + 1 * HH,  xw1, nullptr, Wr2 + 1 * HH,
        acc_wb, cnt_wb, Wl2 + 3 * HH,  xw1, nullptr, Wr2 + 3 * HH,
        bl2 + 1 * H, bl2 + 3 * H, xw2, sw, 4, 0);

    // -------- classifier
    edge_dot<<<cdiv((long)NE_LBL * 32, TB), TB, 0, stream>>>(xw2, xe2, e_lbl, e_lbl + NE_LBL, pred);
}